// NeuralMapCell_39049842655613
// MI455X (gfx1250) — compile-verified
//
#include <hip/hip_runtime.h>
#include <hip/hip_bf16.h>

#define B_      256
#define MEM_    128
#define UNITS_  1024
#define EDIM_   1024
#define INROW_  (EDIM_ + UNITS_)   // 2048

typedef __attribute__((ext_vector_type(2))) float v2f;
typedef __attribute__((ext_vector_type(8))) float v8f;

// ---------------------------------------------------------------------------
// Kernel 1: fused attention (logits -> softmax -> weighted sum) + shift-copy.
// One block per batch element; 256 threads = 8 wave32.
// Streams M_key and M_value from HBM exactly once, writing the shifted copy
// (rows 1..127 -> rows 0..126) on the way through.
// ---------------------------------------------------------------------------
__global__ void __launch_bounds__(256)
attn_shift_kernel(const float* __restrict__ inputs,
                  const float* __restrict__ M_key,
                  const float* __restrict__ M_value,
                  float* __restrict__ out_output,   // (B, UNITS)
                  float* __restrict__ out_Mkey,     // (B, MEM, UNITS)
                  float* __restrict__ out_Mval)     // (B, MEM, UNITS)
{
    __shared__ float sh_h[UNITS_];
    __shared__ float sh_logit[MEM_];
    __shared__ float sh_at[MEM_];
    __shared__ float sh_red[2];

    const int b    = blockIdx.x;
    const int t    = threadIdx.x;
    const int lane = t & 31;
    const int wave = t >> 5;

    // --- stage h_t (inputs[b, EDIM: EDIM+UNITS]) into LDS, coalesced ---
    const float* hrow = inputs + (size_t)b * INROW_ + EDIM_;
#pragma unroll
    for (int j = 0; j < 4; ++j)
        sh_h[t + 256 * j] = hrow[t + 256 * j];
    __syncthreads();

    // --- phase 1: logits[m] = dot(M_key[b,m,:], h) ; shift-copy M_key ---
    // wave w owns rows m = 16w .. 16w+15
    const size_t bbase = (size_t)b * MEM_ * UNITS_;
#pragma unroll 1
    for (int r = 0; r < 16; ++r) {
        const int m = wave * 16 + r;
        const float* row = M_key + bbase + (size_t)m * UNITS_;
        float* dst = out_Mkey + bbase + (size_t)(m - 1) * UNITS_; // valid when m>=1
        float acc = 0.0f;
#pragma unroll 8
        for (int i = 0; i < UNITS_ / 32; ++i) {
            const int u = lane + 32 * i;
            const float v = row[u];
            acc += v * sh_h[u];
            if (m >= 1) dst[u] = v;   // uniform per-row branch, coalesced 128B store
        }
        // wave32 butterfly reduction
#pragma unroll
        for (int off = 16; off >= 1; off >>= 1)
            acc += __shfl_xor(acc, off, 32);
        if (lane == 0) sh_logit[m] = acc;
    }
    __syncthreads();

    // --- softmax over MEM_ = 128 logits (tiny; serial scans are fine) ---
    if (t == 0) {
        float mx = sh_logit[0];
        for (int m = 1; m < MEM_; ++m) mx = fmaxf(mx, sh_logit[m]);
        sh_red[0] = mx;
    }
    __syncthreads();
    if (t < MEM_) sh_at[t] = __expf(sh_logit[t] - sh_red[0]);
    __syncthreads();
    if (t == 0) {
        float s = 0.0f;
        for (int m = 0; m < MEM_; ++m) s += sh_at[m];
        sh_red[1] = 1.0f / s;
    }
    __syncthreads();

    // --- phase 2: output[b,u] = sum_m at[m]*M_value[b,m,u] ; shift-copy ---
    const float inv = sh_red[1];
    float acc0 = 0.f, acc1 = 0.f, acc2 = 0.f, acc3 = 0.f;
#pragma unroll 1
    for (int m = 0; m < MEM_; ++m) {
        const float a = sh_at[m] * inv;
        const float* row = M_value + bbase + (size_t)m * UNITS_;
        float* dst = out_Mval + bbase + (size_t)(m - 1) * UNITS_;
        const float v0 = row[t + 0];
        const float v1 = row[t + 256];
        const float v2 = row[t + 512];
        const float v3 = row[t + 768];
        acc0 += a * v0; acc1 += a * v1; acc2 += a * v2; acc3 += a * v3;
        if (m >= 1) {
            dst[t + 0]   = v0;
            dst[t + 256] = v1;
            dst[t + 512] = v2;
            dst[t + 768] = v3;
        }
    }
    float* orow = out_output + (size_t)b * UNITS_;
    orow[t + 0]   = acc0;
    orow[t + 256] = acc1;
    orow[t + 512] = acc2;
    orow[t + 768] = acc3;
}

// ---------------------------------------------------------------------------
// Kernel 2: m_key = e_t @ W_key and m_value = e_t @ W_value via f32 WMMA,
// written directly into memory slot 127 of new_M_key / new_M_value.
// One 16x16 output tile per wave, K-loop in steps of 4
// (V_WMMA_F32_16X16X4_F32). grid = (Ntiles/8, Mtiles, 2), block = 8 waves.
// ---------------------------------------------------------------------------
__global__ void __launch_bounds__(256)
gemm_tail_kernel(const float* __restrict__ inputs,
                 const float* __restrict__ W_key,
                 const float* __restrict__ W_value,
                 float* __restrict__ out_Mkey,
                 float* __restrict__ out_Mval)
{
    const int lane = threadIdx.x & 31;
    const int wave = threadIdx.x >> 5;

    const int nt = blockIdx.x * 8 + wave;  // 0..63  (N tile)
    const int mt = blockIdx.y;             // 0..15  (M tile == batch tile)
    const float* W  = blockIdx.z ? W_value : W_key;
    float*      out = blockIdx.z ? out_Mval : out_Mkey;

    const int m0 = mt * 16;
    const int n0 = nt * 16;

    // A fragment lane mapping (16x4 f32): M = lane&15, K pair = 2*(lane>>4)
    const int mr = lane & 15;
    const int kb = (lane >> 4) * 2;

    const float* Arow = inputs + (size_t)(m0 + mr) * INROW_;     // e_t row (k < EDIM_)
    const float* Bcol = W + (size_t)n0 + (lane & 15);            // column n0+lane%16, stride UNITS_

    v8f c = {};
#pragma unroll 4
    for (int k = 0; k < EDIM_; k += 4) {
        // A: contiguous float2 at (mr, k+kb)
        v2f a = *(const v2f*)(Arow + k + kb);
        // B: two strided scalars at (k+kb, n), (k+kb+1, n)
        v2f bf;
        bf.x = Bcol[(size_t)(k + kb) * UNITS_];
        bf.y = Bcol[(size_t)(k + kb + 1) * UNITS_];
        c = __builtin_amdgcn_wmma_f32_16x16x4_f32(
                /*neg_a=*/false, a, /*neg_b=*/false, bf,
                /*c_mod=*/(short)0, c, /*reuse_a=*/false, /*reuse_b=*/false);
    }

    // C/D layout: lanes 0-15 -> rows 0..7 in v0..v7, lanes 16-31 -> rows 8..15
    const int rbase = (lane >> 4) * 8;
    const int col   = n0 + (lane & 15);
#pragma unroll
    for (int r = 0; r < 8; ++r) {
        const int row = m0 + rbase + r;  // batch index
        out[((size_t)row * MEM_ + (MEM_ - 1)) * UNITS_ + col] = c[r];
    }
}

extern "C" void kernel_launch(void* const* d_in, const int* in_sizes, int n_in,
                              void* d_out, int out_size, void* d_ws, size_t ws_size,
                              hipStream_t stream) {
    const float* inputs  = (const float*)d_in[0];
    const float* M_key   = (const float*)d_in[1];
    const float* M_value = (const float*)d_in[2];
    const float* W_key   = (const float*)d_in[3];
    const float* W_value = (const float*)d_in[4];

    float* out_output = (float*)d_out;                                   // B*UNITS
    float* out_Mkey   = out_output + (size_t)B_ * UNITS_;                // B*MEM*UNITS
    float* out_Mval   = out_Mkey + (size_t)B_ * MEM_ * UNITS_;           // B*MEM*UNITS

    attn_shift_kernel<<<B_, 256, 0, stream>>>(inputs, M_key, M_value,
                                              out_output, out_Mkey, out_Mval);

    dim3 grid(UNITS_ / 16 / 8, B_ / 16, 2);  // (8, 16, 2)
    gemm_tail_kernel<<<grid, 256, 0, stream>>>(inputs, W_key, W_value,
                                               out_Mkey, out_Mval);
}